// GINConvNet_31988916420624
// MI455X (gfx1250) — compile-verified
//
#include <hip/hip_runtime.h>
#include <math.h>

// ---------------------------------------------------------------------------
// GIN (2x GINConv + mean-pool + sigmoid head) for MI455X / gfx1250.
// Edge aggregation is memory/atomic bound (L2-resident); node MLPs use
// V_WMMA_F32_16X16X4_F32 (full fp32 tensor path, 16-node tiles per wave32).
// ---------------------------------------------------------------------------

typedef __attribute__((ext_vector_type(2))) float v2f;
typedef __attribute__((ext_vector_type(8))) float v8f;

__global__ void k_zero(float* __restrict__ p, int n) {
  int i = blockIdx.x * blockDim.x + threadIdx.x;
  if (i < n) p[i] = 0.0f;
}

// Layer-1 aggregation: d=1 scalar gather/scatter over edges.
__global__ void k_agg1(const float* __restrict__ x,
                       const int* __restrict__ src,
                       const int* __restrict__ dst,
                       float* __restrict__ agg1, int E) {
  int e = blockIdx.x * blockDim.x + threadIdx.x;
  if (e < E) atomicAdd(&agg1[dst[e]], x[src[e]]);
}

// Layer-2 aggregation: 8 threads per edge (one per feature).
__global__ void k_agg2(const float* __restrict__ h1,
                       const int* __restrict__ src,
                       const int* __restrict__ dst,
                       float* __restrict__ agg2, int E) {
  int i = blockIdx.x * blockDim.x + threadIdx.x;
  if (i < E * 8) {
    int e = i >> 3, f = i & 7;
    atomicAdd(&agg2[dst[e] * 8 + f], h1[src[e] * 8 + f]);
  }
}

// Layer 1: per wave, 16 nodes.  s = x + agg1 (scalar feature);
// t = relu(s*W1a + b1a) [16x8]; h1 = elu(t @ W1b + b1b) via two
// V_WMMA_F32_16X16X4_F32 (K=8 split into 2 chunks), bias in initial C.
__global__ void k_layer1(const float* __restrict__ x,
                         const float* __restrict__ agg1,
                         const float* __restrict__ W1a, const float* __restrict__ b1a,
                         const float* __restrict__ W1b, const float* __restrict__ b1b,
                         float* __restrict__ h1, int N, int ntiles) {
  int gtid = blockIdx.x * blockDim.x + threadIdx.x;
  int tile = gtid >> 5;
  if (tile >= ntiles) return;            // wave-uniform exit: EXEC stays all-1s
  int lane = gtid & 31;
  int row  = lane & 15;                  // A-matrix row  /  B,C,D column n
  int half = lane >> 4;                  // K-half for A/B, M-half for C/D
  int node_a = tile * 16 + row;
  if (node_a >= N) node_a = N - 1;       // clamp (junk rows never stored)
  float s = x[node_a] + agg1[node_a];

  int k0 = 2 * half;                     // K indices: chunk0 {k0,k0+1}, chunk1 +4
  v2f a0, a1, b0, b1;
  a0.x = fmaxf(0.0f, fmaf(s, W1a[k0 + 0], b1a[k0 + 0]));
  a0.y = fmaxf(0.0f, fmaf(s, W1a[k0 + 1], b1a[k0 + 1]));
  a1.x = fmaxf(0.0f, fmaf(s, W1a[k0 + 4], b1a[k0 + 4]));
  a1.y = fmaxf(0.0f, fmaf(s, W1a[k0 + 5], b1a[k0 + 5]));

  bool vn = row < 8;                     // valid output feature columns 0..7
  b0.x = vn ? W1b[(k0 + 0) * 8 + row] : 0.0f;
  b0.y = vn ? W1b[(k0 + 1) * 8 + row] : 0.0f;
  b1.x = vn ? W1b[(k0 + 4) * 8 + row] : 0.0f;
  b1.y = vn ? W1b[(k0 + 5) * 8 + row] : 0.0f;

  v8f c;
  float cb = vn ? b1b[row] : 0.0f;       // bias through initial accumulator
  #pragma unroll
  for (int r = 0; r < 8; ++r) c[r] = cb;

  c = __builtin_amdgcn_wmma_f32_16x16x4_f32(false, a0, false, b0, (short)0, c, false, false);
  c = __builtin_amdgcn_wmma_f32_16x16x4_f32(false, a1, false, b1, (short)0, c, false, false);

  if (vn) {
    #pragma unroll
    for (int r = 0; r < 8; ++r) {
      int node_d = tile * 16 + half * 8 + r;   // D row m = r + 8*half
      if (node_d < N) {
        float v = c[r];
        v = (v > 0.0f) ? v : expm1f(v);        // ELU (alpha=1)
        h1[node_d * 8 + row] = v;
      }
    }
  }
}

// Layer 2 fused with mean-pool accumulation.
// z = h1 + agg2; t = relu(z @ W2a + b2a) computed per-lane (avoids D->A
// transpose); h2 = t @ W2b + b2b via two f32 WMMAs; atomically pooled.
__global__ void k_layer2_pool(const float* __restrict__ h1,
                              const float* __restrict__ agg2,
                              const float* __restrict__ W2a, const float* __restrict__ b2a,
                              const float* __restrict__ W2b, const float* __restrict__ b2b,
                              const int* __restrict__ batch,
                              float* __restrict__ gsum, int N, int ntiles) {
  int gtid = blockIdx.x * blockDim.x + threadIdx.x;
  int tile = gtid >> 5;
  if (tile >= ntiles) return;
  int lane = gtid & 31;
  int row  = lane & 15;
  int half = lane >> 4;
  int node_a = tile * 16 + row;
  if (node_a >= N) node_a = N - 1;

  const float4* hp = (const float4*)(h1   + node_a * 8);
  const float4* ap = (const float4*)(agg2 + node_a * 8);
  float4 hA = hp[0], hB = hp[1], qA = ap[0], qB = ap[1];
  float z[8];
  z[0] = hA.x + qA.x; z[1] = hA.y + qA.y; z[2] = hA.z + qA.z; z[3] = hA.w + qA.w;
  z[4] = hB.x + qB.x; z[5] = hB.y + qB.y; z[6] = hB.z + qB.z; z[7] = hB.w + qB.w;

  int k0 = 2 * half;
  // t[k] = relu(sum_j z[j] * W2a[j][k] + b2a[k]) for the 4 K slots this lane feeds
  float t00 = b2a[k0 + 0], t01 = b2a[k0 + 1], t10 = b2a[k0 + 4], t11 = b2a[k0 + 5];
  #pragma unroll
  for (int j = 0; j < 8; ++j) {
    const float* wr = W2a + j * 8;
    t00 = fmaf(z[j], wr[k0 + 0], t00);
    t01 = fmaf(z[j], wr[k0 + 1], t01);
    t10 = fmaf(z[j], wr[k0 + 4], t10);
    t11 = fmaf(z[j], wr[k0 + 5], t11);
  }
  v2f a0, a1, b0, b1;
  a0.x = fmaxf(0.0f, t00); a0.y = fmaxf(0.0f, t01);
  a1.x = fmaxf(0.0f, t10); a1.y = fmaxf(0.0f, t11);

  bool vn = row < 8;
  b0.x = vn ? W2b[(k0 + 0) * 8 + row] : 0.0f;
  b0.y = vn ? W2b[(k0 + 1) * 8 + row] : 0.0f;
  b1.x = vn ? W2b[(k0 + 4) * 8 + row] : 0.0f;
  b1.y = vn ? W2b[(k0 + 5) * 8 + row] : 0.0f;

  v8f c;
  float cb = vn ? b2b[row] : 0.0f;
  #pragma unroll
  for (int r = 0; r < 8; ++r) c[r] = cb;

  c = __builtin_amdgcn_wmma_f32_16x16x4_f32(false, a0, false, b0, (short)0, c, false, false);
  c = __builtin_amdgcn_wmma_f32_16x16x4_f32(false, a1, false, b1, (short)0, c, false, false);

  if (vn) {
    #pragma unroll
    for (int r = 0; r < 8; ++r) {
      int node_d = tile * 16 + half * 8 + r;
      if (node_d < N) {
        int g = batch[node_d];
        atomicAdd(&gsum[g * 8 + row], c[r]);   // pooled sum, feature = row
      }
    }
  }
}

__global__ void k_counts(const int* __restrict__ batch, float* __restrict__ gcnt, int N) {
  int i = blockIdx.x * blockDim.x + threadIdx.x;
  if (i < N) atomicAdd(&gcnt[batch[i]], 1.0f);
}

__global__ void k_final(const float* __restrict__ gsum, const float* __restrict__ gcnt,
                        const float* __restrict__ Wfc, const float* __restrict__ bfc,
                        float* __restrict__ out, int G) {
  int g = blockIdx.x * blockDim.x + threadIdx.x;
  if (g < G) {
    float cnt = fmaxf(gcnt[g], 1.0f);
    float acc = bfc[0];
    #pragma unroll
    for (int j = 0; j < 8; ++j) acc = fmaf(gsum[g * 8 + j] / cnt, Wfc[j], acc);
    out[g] = 1.0f / (1.0f + expf(-acc));
  }
}

extern "C" void kernel_launch(void* const* d_in, const int* in_sizes, int n_in,
                              void* d_out, int out_size, void* d_ws, size_t ws_size,
                              hipStream_t stream) {
  const float* x          = (const float*)d_in[0];
  const int*   edge_index = (const int*)  d_in[1];
  const int*   batch      = (const int*)  d_in[2];
  const float* W1a = (const float*)d_in[3];
  const float* b1a = (const float*)d_in[4];
  const float* W1b = (const float*)d_in[5];
  const float* b1b = (const float*)d_in[6];
  const float* W2a = (const float*)d_in[7];
  const float* b2a = (const float*)d_in[8];
  const float* W2b = (const float*)d_in[9];
  const float* b2b = (const float*)d_in[10];
  const float* Wfc = (const float*)d_in[11];
  const float* bfc = (const float*)d_in[12];
  float* out = (float*)d_out;

  const int N = in_sizes[0];        // 200000
  const int E = in_sizes[1] / 2;    // 6400000
  const int G = out_size;           // 1024
  const int* src = edge_index;      // edge_index[0, :]
  const int* dst = edge_index + E;  // edge_index[1, :]

  // Workspace layout (floats): agg1[N] | agg2[8N] | gsum[8G] | gcnt[G] | h1[8N]
  float* ws   = (float*)d_ws;
  float* agg1 = ws;
  float* agg2 = agg1 + N;
  float* gsum = agg2 + 8 * N;
  float* gcnt = gsum + 8 * G;
  float* h1   = gcnt + G;

  const int BLK = 256;
  // Zero all accumulators (agg1, agg2, gsum, gcnt) every call: deterministic.
  int nz = 9 * N + 9 * G;
  k_zero<<<(nz + BLK - 1) / BLK, BLK, 0, stream>>>(ws, nz);

  k_agg1<<<(E + BLK - 1) / BLK, BLK, 0, stream>>>(x, src, dst, agg1, E);

  int ntiles = (N + 15) / 16;
  int nthr   = ntiles * 32;
  k_layer1<<<(nthr + BLK - 1) / BLK, BLK, 0, stream>>>(x, agg1, W1a, b1a, W1b, b1b,
                                                       h1, N, ntiles);

  int n2 = E * 8;
  k_agg2<<<(n2 + BLK - 1) / BLK, BLK, 0, stream>>>(h1, src, dst, agg2, E);

  k_layer2_pool<<<(nthr + BLK - 1) / BLK, BLK, 0, stream>>>(h1, agg2, W2a, b2a, W2b, b2b,
                                                            batch, gsum, N, ntiles);

  k_counts<<<(N + BLK - 1) / BLK, BLK, 0, stream>>>(batch, gcnt, N);
  k_final<<<(G + BLK - 1) / BLK, BLK, 0, stream>>>(gsum, gcnt, Wfc, bfc, out, G);
}